// AggLayer_59090160058831
// MI455X (gfx1250) — compile-verified
//
#include <hip/hip_runtime.h>
#include <stdint.h>

#define EPS 1e-5f

typedef __attribute__((ext_vector_type(2))) float v2f;
typedef __attribute__((ext_vector_type(8))) float v8f;

struct SelState { unsigned prefix; int need; };

// ---------------- init: agg[n][o] = bias[o] + sum_i h[n][i]*root[i][o] ----------------
__global__ void gnn_init(const float* __restrict__ hin, int hstride, int in_ch,
                         const float* __restrict__ root, const float* __restrict__ bias,
                         float* __restrict__ agg, int N)
{
    int n = blockIdx.x * blockDim.x + threadIdx.x;
    if (n >= N) return;
    const float* hr = hin + (size_t)n * hstride;
    #pragma unroll
    for (int o = 0; o < 8; ++o) {
        float acc = bias[o];
        for (int i = 0; i < in_ch; ++i) acc += hr[i] * root[i * 8 + o];
        agg[(size_t)n * 8 + o] = acc;
    }
}

// ---------------- nc_in edge kernel (in_ch=1), MLP 2->4->16->8, VALU ----------------
__global__ __launch_bounds__(256) void gnn_edge_in(
    const float* __restrict__ x,
    const int* __restrict__ src, const int* __restrict__ dst,
    const float* __restrict__ attr,
    const float* __restrict__ W0, const float* __restrict__ b0,   // [2,4],[4]
    const float* __restrict__ W1, const float* __restrict__ b1,   // [4,16],[16]
    const float* __restrict__ W2, const float* __restrict__ b2,   // [16,8],[8]
    float* __restrict__ agg, int E)
{
    __shared__ float sm[8 + 4 + 64 + 16 + 128 + 8]; // 228
    float* sW0 = sm;        float* sb0 = sm + 8;
    float* sW1 = sm + 12;   float* sb1 = sm + 76;
    float* sW2 = sm + 92;   float* sb2 = sm + 220;
    int tid = threadIdx.x;
    if (tid < 8)   sW0[tid] = W0[tid];
    if (tid < 4)   sb0[tid] = b0[tid];
    if (tid < 64)  sW1[tid] = W1[tid];
    if (tid < 16)  sb1[tid] = b1[tid];
    if (tid < 128) sW2[tid] = W2[tid];
    if (tid < 8)   sb2[tid] = b2[tid];
    __syncthreads();

    int e = blockIdx.x * 256 + tid;
    if (e >= E) return;
    float a0 = attr[2 * (size_t)e], a1 = attr[2 * (size_t)e + 1];
    float m1[4];
    #pragma unroll
    for (int j = 0; j < 4; ++j) m1[j] = fmaxf(a0 * sW0[j] + a1 * sW0[4 + j] + sb0[j], 0.f);
    float m2[16];
    #pragma unroll
    for (int j = 0; j < 16; ++j) {
        float acc = sb1[j];
        #pragma unroll
        for (int i = 0; i < 4; ++i) acc += m1[i] * sW1[i * 16 + j];
        m2[j] = fmaxf(acc, 0.f);
    }
    float xs = x[src[e]];
    int d = dst[e];
    #pragma unroll
    for (int o = 0; o < 8; ++o) {
        float acc = sb2[o];
        #pragma unroll
        for (int i = 0; i < 16; ++i) acc += m2[i] * sW2[i * 8 + o];
        float w = fmaxf(acc, 0.f);
        atomicAdd(&agg[(size_t)d * 8 + o], xs * w);
    }
}

// ---------------- WMMA edge kernel for 8->8 convs, MLP 2->8->16->64 ----------------
// Per wave: 16 edges/iter. Layer 1 in VALU, layer 2 computed per-lane ONLY for the
// 8 K-columns this lane feeds into the A fragments (runtime column index is used
// solely as an LDS address -> no register-array dynamic indexing, no cndmask chains).
// Layer 3 via v_wmma_f32_16x16x4_f32 (4 N-tiles x 4 K-chunks).
__global__ __launch_bounds__(256) void gnn_edge_wmma(
    const float* __restrict__ hin,                                 // stride-48 slice base
    const int* __restrict__ src, const int* __restrict__ dst,
    const float* __restrict__ attr,
    const float* __restrict__ W1, const float* __restrict__ b1,    // [2,8],[8]
    const float* __restrict__ W2, const float* __restrict__ b2,    // [8,16],[16]
    const float* __restrict__ W3, const float* __restrict__ b3,    // [16,64],[64]
    float* __restrict__ agg, int E)
{
    __shared__ float sm[16 + 8 + 128 + 16]; // 168: W1,b1,W2,b2
    float* sW1 = sm;       float* sb1 = sm + 16;
    float* sW2 = sm + 24;  float* sb2 = sm + 152;
    {
        int tid = threadIdx.x;
        if (tid < 16)  sW1[tid] = W1[tid];
        if (tid < 8)   sb1[tid] = b1[tid];
        if (tid < 128) sW2[tid] = W2[tid];
        if (tid < 16)  sb2[tid] = b2[tid];
    }
    __syncthreads();

    const int  lane = threadIdx.x & 31;
    const int  m    = lane & 15;        // row (edge within group) for A; column for B/D
    const bool hi   = (lane >= 16);     // upper half supplies K-slots 2,3 of each chunk
    const int  koff = hi ? 2 : 0;
    const int  gwave  = blockIdx.x * 8 + (threadIdx.x >> 5);
    const int  nwaves = gridDim.x * 8;

    // Preload B fragments (weights, loop-invariant): B[4x16] chunk cc of W3[16,64], tile t.
    v2f   Bf[4][4];
    float bias3[4];
    #pragma unroll
    for (int t = 0; t < 4; ++t) {
        int col = 16 * t + m;
        bias3[t] = b3[col];
        #pragma unroll
        for (int cc = 0; cc < 4; ++cc) {
            int r0 = 4 * cc + koff;
            v2f bfrag = { W3[r0 * 64 + col], W3[(r0 + 1) * 64 + col] };
            Bf[t][cc] = bfrag;
        }
    }

    const int ngroups = (E + 15) >> 4;
    for (int g = gwave; g < ngroups; g += nwaves) {
        int  e  = g * 16 + m;
        bool ev = (e < E);
        float a0 = ev ? attr[2 * (size_t)e]     : 0.f;
        float a1 = ev ? attr[2 * (size_t)e + 1] : 0.f;
        float t1[8];
        #pragma unroll
        for (int j = 0; j < 8; ++j) t1[j] = fmaxf(a0 * sW1[j] + a1 * sW1[8 + j] + sb1[j], 0.f);
        // Layer 2, only the columns this lane owns: j = 4*cc + koff + u (LDS-indexed).
        v2f Af[4];
        #pragma unroll
        for (int cc = 0; cc < 4; ++cc) {
            float pr[2];
            #pragma unroll
            for (int u = 0; u < 2; ++u) {
                int j = 4 * cc + koff + u;      // runtime, but only an LDS address
                float acc = sb2[j];
                #pragma unroll
                for (int i = 0; i < 8; ++i) acc += t1[i] * sW2[i * 16 + j];
                pr[u] = fmaxf(acc, 0.f);
            }
            v2f af = { pr[0], pr[1] };
            Af[cc] = af;
        }
        // Layer 3: [16 edges x 16] @ [16 x 64] as 4 N-tiles x 4 K-chunks of 16x16x4 fp32 WMMA
        v8f D[4];
        #pragma unroll
        for (int t = 0; t < 4; ++t) {
            v8f acc = { 0.f, 0.f, 0.f, 0.f, 0.f, 0.f, 0.f, 0.f };
            #pragma unroll
            for (int cc = 0; cc < 4; ++cc) {
                acc = __builtin_amdgcn_wmma_f32_16x16x4_f32(
                    false, Af[cc], false, Bf[t][cc], (short)0, acc, false, false);
            }
            D[t] = acc;
        }
        // msg[edge][o] += sum_i x_src[i] * relu(D + bias). Lane owns (i = 2t+h, o = lane&7).
        int rowbase = g * 16 + (hi ? 8 : 0); // D VGPR r -> row r or r+8
        int o = lane & 7;
        int h = m >> 3;                      // global-pointer offset, pure address math
        float part[8];
        #pragma unroll
        for (int r = 0; r < 8; ++r) {
            int er = rowbase + r;
            float p = 0.f;
            if (er < E) {
                const float* xr = hin + (size_t)src[er] * 48 + h;
                #pragma unroll
                for (int t = 0; t < 4; ++t) {
                    float w = fmaxf(D[t][r] + bias3[t], 0.f);
                    p += w * xr[2 * t];
                }
            }
            part[r] = p;
        }
        #pragma unroll
        for (int r = 0; r < 8; ++r) part[r] += __shfl_xor(part[r], 8, 32);
        if ((lane & 8) == 0) {
            #pragma unroll
            for (int r = 0; r < 8; ++r) {
                int er = rowbase + r;
                if (er < E) atomicAdd(&agg[(size_t)dst[er] * 8 + o], part[r]);
            }
        }
    }
}

// ---------------- relu + per-block channel stats, write feats slice ----------------
__global__ __launch_bounds__(256) void gnn_relustats(const float* __restrict__ agg,
    float* __restrict__ outSlice, float* __restrict__ blockSums, int N)
{
    __shared__ float ssum[256 * 8];
    __shared__ float ssq[256 * 8];
    int tid = threadIdx.x;
    int n = blockIdx.x * 256 + tid;
    float y[8];
    if (n < N) {
        #pragma unroll
        for (int c = 0; c < 8; ++c) {
            y[c] = fmaxf(agg[(size_t)n * 8 + c], 0.f);
            outSlice[(size_t)n * 48 + c] = y[c];
        }
    } else {
        #pragma unroll
        for (int c = 0; c < 8; ++c) y[c] = 0.f;
    }
    #pragma unroll
    for (int c = 0; c < 8; ++c) { ssum[tid * 8 + c] = y[c]; ssq[tid * 8 + c] = y[c] * y[c]; }
    __syncthreads();
    for (int s = 128; s > 0; s >>= 1) {
        if (tid < s) {
            #pragma unroll
            for (int c = 0; c < 8; ++c) {
                ssum[tid * 8 + c] += ssum[(tid + s) * 8 + c];
                ssq[tid * 8 + c]  += ssq[(tid + s) * 8 + c];
            }
        }
        __syncthreads();
    }
    if (tid < 8)            blockSums[blockIdx.x * 16 + tid] = ssum[tid];
    else if (tid < 16)      blockSums[blockIdx.x * 16 + tid] = ssq[tid - 8];
}

// deterministic final reduction of per-block stats
__global__ void gnn_stats2(const float* __restrict__ blockSums, float* __restrict__ stats, int nb)
{
    int t = threadIdx.x;
    if (t >= 16) return;
    float acc = 0.f;
    for (int b = 0; b < nb; ++b) acc += blockSums[b * 16 + t];
    stats[t] = acc;
}

__global__ void gnn_norm(float* __restrict__ slice, const float* __restrict__ stats, int N)
{
    int n = blockIdx.x * blockDim.x + threadIdx.x;
    if (n >= N) return;
    float invN = 1.f / (float)N;
    #pragma unroll
    for (int c = 0; c < 8; ++c) {
        float mean = stats[c] * invN;
        float var  = stats[8 + c] * invN - mean * mean;
        float v = slice[(size_t)n * 48 + c];
        slice[(size_t)n * 48 + c] = (v - mean) * rsqrtf(var + EPS);
    }
}

// ---------------- final MLP 48->24->16->8->1 (ReLU after every layer) ----------------
#define FW0 0
#define FB0 1152
#define FW1 1176
#define FB1 1560
#define FW2 1576
#define FB2 1704
#define FW3 1712
#define FB3 1720
#define FTOT 1721

__global__ __launch_bounds__(256) void gnn_fmap(const float* __restrict__ feats,
    const float* __restrict__ packed, float* __restrict__ sbuf, float* __restrict__ out, int N)
{
    __shared__ float w[FTOT];
    for (int i = threadIdx.x; i < FTOT; i += 256) w[i] = packed[i];
    __syncthreads();
    int n = blockIdx.x * 256 + threadIdx.x;
    if (n >= N) return;
    float f0[48];
    #pragma unroll
    for (int i = 0; i < 48; ++i) f0[i] = feats[(size_t)n * 48 + i];
    float f1[24];
    for (int j = 0; j < 24; ++j) {
        float acc = w[FB0 + j];
        #pragma unroll
        for (int i = 0; i < 48; ++i) acc += f0[i] * w[FW0 + i * 24 + j];
        f1[j] = fmaxf(acc, 0.f);
    }
    float f2[16];
    for (int j = 0; j < 16; ++j) {
        float acc = w[FB1 + j];
        #pragma unroll
        for (int i = 0; i < 24; ++i) acc += f1[i] * w[FW1 + i * 16 + j];
        f2[j] = fmaxf(acc, 0.f);
    }
    float f3[8];
    #pragma unroll
    for (int j = 0; j < 8; ++j) {
        float acc = w[FB2 + j];
        #pragma unroll
        for (int i = 0; i < 16; ++i) acc += f2[i] * w[FW2 + i * 8 + j];
        f3[j] = fmaxf(acc, 0.f);
    }
    float s = w[FB3];
    #pragma unroll
    for (int i = 0; i < 8; ++i) s += f3[i] * w[FW3 + i];
    s = fmaxf(s, 0.f);
    sbuf[n] = s;
    out[2 * (size_t)n] = s;
}

// ---------------- radix-select (s >= 0 so float bits are order-preserving) ----------------
__global__ void gnn_radix_init(unsigned* __restrict__ hist, SelState* st, const int* __restrict__ dk)
{
    int t = threadIdx.x;
    hist[t] = 0u;
    if (t == 0) { st->prefix = 0u; st->need = *dk; }
}

__global__ void gnn_hist(const float* __restrict__ sbuf, unsigned* __restrict__ hist,
                         const SelState* st, int N, int shift)
{
    int n0 = blockIdx.x * blockDim.x + threadIdx.x;
    int stride = gridDim.x * blockDim.x;
    unsigned pfx = st->prefix;
    for (int n = n0; n < N; n += stride) {
        unsigned u = __float_as_uint(sbuf[n]);
        bool ok = (shift == 24) || ((u >> (shift + 8)) == (pfx >> (shift + 8)));
        if (ok) atomicAdd(&hist[(u >> shift) & 255u], 1u);
    }
}

__global__ void gnn_digit(unsigned* __restrict__ hist, SelState* st, int shift)
{
    if (threadIdx.x == 0) {
        int need = st->need;
        unsigned cum = 0; int d = 255;
        for (; d > 0; --d) {
            if (cum + hist[d] >= (unsigned)need) break;
            cum += hist[d];
        }
        st->prefix |= ((unsigned)d) << shift;
        st->need = need - (int)cum;
    }
    __syncthreads();
    hist[threadIdx.x] = 0u;
}

__global__ __launch_bounds__(256) void gnn_selcount(const float* __restrict__ sbuf,
    const SelState* st, float* __restrict__ out, unsigned* __restrict__ blockTie, int N)
{
    __shared__ unsigned cnt[256];
    int tid = threadIdx.x;
    int n = blockIdx.x * 256 + tid;
    unsigned thr = st->prefix;
    int tie = 0;
    if (n < N) {
        unsigned u = __float_as_uint(sbuf[n]);
        tie = (u == thr);
        out[2 * (size_t)n + 1] = (u > thr) ? 1.f : 0.f;
    }
    cnt[tid] = (unsigned)tie;
    __syncthreads();
    for (int s = 128; s > 0; s >>= 1) {
        if (tid < s) cnt[tid] += cnt[tid + s];
        __syncthreads();
    }
    if (tid == 0) blockTie[blockIdx.x] = cnt[0];
}

__global__ void gnn_scan(const unsigned* __restrict__ blockTie,
                         unsigned* __restrict__ blockTieOff, int nb)
{
    if (threadIdx.x == 0) {
        unsigned run = 0;
        for (int b = 0; b < nb; ++b) { blockTieOff[b] = run; run += blockTie[b]; }
    }
}

__global__ __launch_bounds__(256) void gnn_selmark(const float* __restrict__ sbuf,
    const SelState* st, const unsigned* __restrict__ blockTieOff,
    float* __restrict__ out, int N)
{
    __shared__ int sc[256];
    int tid = threadIdx.x;
    int n = blockIdx.x * 256 + tid;
    unsigned thr = st->prefix;
    int need = st->need;
    int tie = 0;
    if (n < N) tie = (__float_as_uint(sbuf[n]) == thr);
    sc[tid] = tie;
    __syncthreads();
    for (int ofs = 1; ofs < 256; ofs <<= 1) {
        int v = sc[tid];
        if (tid >= ofs) v += sc[tid - ofs];
        __syncthreads();
        sc[tid] = v;
        __syncthreads();
    }
    if (n < N && tie) {
        int rank = (int)blockTieOff[blockIdx.x] + sc[tid] - 1;
        out[2 * (size_t)n + 1] = (rank < need) ? 1.f : 0.f;
    }
}

// ---------------- host launcher ----------------
extern "C" void kernel_launch(void* const* d_in, const int* in_sizes, int n_in,
                              void* d_out, int out_size, void* d_ws, size_t ws_size,
                              hipStream_t stream)
{
    const float* x    = (const float*)d_in[0];
    const int*   eidx = (const int*)d_in[1];
    const float* attr = (const float*)d_in[2];
    const int*   dk   = (const int*)d_in[3];
    const int N = in_sizes[0];
    const int E = in_sizes[1] / 2;
    const int* src = eidx;
    const int* dst = eidx + E;

    // logical param slots: 0..7 nc_in{m0w,m0b,m1w,m1b,m2w,m2b,root,bias},
    // 8+8c..: ncs[c] same order, 48..55 fmap{W0,B0,W1,B1,W2,B2,W3,B3}
    const float* P[56];
    int idx = 4;
    if (in_sizes[4] == 24) {
        // jax sorted-key tree order: fmap, nc_in, ncs; dicts sorted (b<w, bias<mlp<root)
        static const int fmapA[8] = {49, 48, 51, 50, 53, 52, 55, 54};
        for (int i = 0; i < 8; ++i) P[fmapA[i]] = (const float*)d_in[idx++];
        static const int ncA[8] = {7, 1, 0, 3, 2, 5, 4, 6}; // bias,m0b,m0w,m1b,m1w,m2b,m2w,root
        for (int i = 0; i < 8; ++i) P[ncA[i]] = (const float*)d_in[idx++];
        for (int c = 0; c < 5; ++c)
            for (int i = 0; i < 8; ++i) P[8 + 8 * c + ncA[i]] = (const float*)d_in[idx++];
    } else {
        // insertion order: nc_in{mlp(w,b)x3,root,bias}, ncs, fmap{(w,b)x4}
        for (int i = 0; i < 8; ++i) P[i] = (const float*)d_in[idx++];
        for (int c = 0; c < 5; ++c)
            for (int i = 0; i < 8; ++i) P[8 + 8 * c + i] = (const float*)d_in[idx++];
        for (int i = 0; i < 8; ++i) P[48 + i] = (const float*)d_in[idx++];
    }

    // workspace carve
    char* ws = (char*)d_ws;
    size_t off = 0;
    auto carve = [&](size_t bytes) -> void* {
        void* p = ws + off;
        off = (off + bytes + 255) & ~(size_t)255;
        return p;
    };
    const int nb = (N + 255) / 256;
    float*    feats       = (float*)carve((size_t)N * 48 * sizeof(float));
    float*    agg         = (float*)carve((size_t)N * 8 * sizeof(float));
    float*    sbuf        = (float*)carve((size_t)N * sizeof(float));
    float*    fpk         = (float*)carve(FTOT * sizeof(float));
    float*    blockSums   = (float*)carve((size_t)nb * 16 * sizeof(float));
    unsigned* blockTie    = (unsigned*)carve((size_t)nb * sizeof(unsigned));
    unsigned* blockTieOff = (unsigned*)carve((size_t)nb * sizeof(unsigned));
    float*    stats       = (float*)carve(16 * sizeof(float));
    unsigned* hist        = (unsigned*)carve(256 * sizeof(unsigned));
    SelState* st          = (SelState*)carve(sizeof(SelState));
    float*    out         = (float*)d_out;

    // pack fmap weights contiguously (d2d, graph-capture safe)
    hipMemcpyAsync(fpk + FW0, P[48], 1152 * 4, hipMemcpyDeviceToDevice, stream);
    hipMemcpyAsync(fpk + FB0, P[49],   24 * 4, hipMemcpyDeviceToDevice, stream);
    hipMemcpyAsync(fpk + FW1, P[50],  384 * 4, hipMemcpyDeviceToDevice, stream);
    hipMemcpyAsync(fpk + FB1, P[51],   16 * 4, hipMemcpyDeviceToDevice, stream);
    hipMemcpyAsync(fpk + FW2, P[52],  128 * 4, hipMemcpyDeviceToDevice, stream);
    hipMemcpyAsync(fpk + FB2, P[53],    8 * 4, hipMemcpyDeviceToDevice, stream);
    hipMemcpyAsync(fpk + FW3, P[54],    8 * 4, hipMemcpyDeviceToDevice, stream);
    hipMemcpyAsync(fpk + FB3, P[55],    1 * 4, hipMemcpyDeviceToDevice, stream);

    // ---- conv 0 (nc_in): x -> feats slice 0 ----
    gnn_init<<<nb, 256, 0, stream>>>(x, 1, 1, P[6], P[7], agg, N);
    gnn_edge_in<<<(E + 255) / 256, 256, 0, stream>>>(x, src, dst, attr,
        P[0], P[1], P[2], P[3], P[4], P[5], agg, E);
    gnn_relustats<<<nb, 256, 0, stream>>>(agg, feats + 0, blockSums, N);
    gnn_stats2<<<1, 32, 0, stream>>>(blockSums, stats, nb);
    gnn_norm<<<nb, 256, 0, stream>>>(feats + 0, stats, N);

    // ---- convs 1..5 (WMMA edge MLP): slice c -> slice c+1 ----
    for (int c = 0; c < 5; ++c) {
        const float** B = &P[8 + 8 * c];
        float* hin = feats + c * 8;
        gnn_init<<<nb, 256, 0, stream>>>(hin, 48, 8, B[6], B[7], agg, N);
        gnn_edge_wmma<<<768, 256, 0, stream>>>(hin, src, dst, attr,
            B[0], B[1], B[2], B[3], B[4], B[5], agg, E);
        gnn_relustats<<<nb, 256, 0, stream>>>(agg, feats + (c + 1) * 8, blockSums, N);
        gnn_stats2<<<1, 32, 0, stream>>>(blockSums, stats, nb);
        gnn_norm<<<nb, 256, 0, stream>>>(feats + (c + 1) * 8, stats, N);
    }

    // ---- final MLP ----
    gnn_fmap<<<nb, 256, 0, stream>>>(feats, fpk, sbuf, out, N);

    // ---- top-k via 4-pass radix select on float bits ----
    gnn_radix_init<<<1, 256, 0, stream>>>(hist, st, dk);
    for (int p = 0; p < 4; ++p) {
        int shift = 24 - 8 * p;
        gnn_hist<<<256, 256, 0, stream>>>(sbuf, hist, st, N, shift);
        gnn_digit<<<1, 256, 0, stream>>>(hist, st, shift);
    }
    gnn_selcount<<<nb, 256, 0, stream>>>(sbuf, st, out, blockTie, N);
    gnn_scan<<<1, 32, 0, stream>>>(blockTie, blockTieOff, nb);
    gnn_selmark<<<nb, 256, 0, stream>>>(sbuf, st, blockTieOff, out, N);
}